// SeathruNeRFRGBRenderer_16381005267471
// MI455X (gfx1250) — compile-verified
//
#include <hip/hip_runtime.h>
#include <math.h>
#include <stdint.h>

// ---------------------------------------------------------------------------
// SeathruNeRF RGB renderer for MI455X (gfx1250).
// Memory-bound (~286 MB traffic -> ~12us floor @ 23.3 TB/s). No matmul -> no
// WMMA; the CDNA5 feature that matters is async global->LDS streaming
// (global_load_async_to_lds_b128 / ASYNCcnt) with double buffering, which
// converts the 3KB-strided per-ray access pattern into dense coalesced
// 16B/lane bulk loads. Inputs are single-pass streams -> non-temporal loads;
// per-sample outputs are write-once -> non-temporal b128 stores.
// ---------------------------------------------------------------------------

namespace {
constexpr int S       = 256;   // samples per ray
constexpr int TPB     = 64;    // threads (=rays) per block, 2 wave32
constexpr int CHUNK   = 8;     // samples streamed per stage
constexpr int NCHUNK  = S / CHUNK;  // 32
// LDS strides in floats. STRIDE3: 24 data + 4 pad (16B-aligned b128 writes,
// gcd(28,64)=4 -> 2-way read conflicts). STRIDE1: 8 floats, no pad
// (gcd(8,64)=8 -> 4-way conflicts on only 2 of 14 reads/sample; keeps the
// whole double buffer at exactly 64KB).
constexpr int STRIDE3 = 28;
constexpr int STRIDE1 = 8;
constexpr int OFF_OBJ = 0;
constexpr int OFF_MED = OFF_OBJ + TPB * STRIDE3;
constexpr int OFF_DIR = OFF_MED + TPB * STRIDE3;
constexpr int OFF_BSC = OFF_DIR + TPB * STRIDE3;
constexpr int OFF_DEN = OFF_BSC + TPB * STRIDE3;
constexpr int OFF_DEL = OFF_DEN + TPB * STRIDE1;
constexpr int BUF_FLOATS = OFF_DEL + TPB * STRIDE1;   // 8192 floats = 32 KB
}  // namespace

typedef float v4f __attribute__((ext_vector_type(4)));

__device__ __forceinline__ float nn(float x) {        // jnp.nan_to_num
  if (x != x) return 0.0f;
  return fminf(fmaxf(x, -3.402823466e38f), 3.402823466e38f);
}
__device__ __forceinline__ float clamp01(float x) {
  return fminf(fmaxf(x, 0.0f), 1.0f);
}

// Async 16B global -> LDS copy, non-temporal (single-pass stream). VDST is
// the wave-relative LDS byte offset (low 32 bits of the generic shared
// pointer), VADDR the 64-bit global address. Tracked by ASYNCcnt.
__device__ __forceinline__ void async_b128(const float* g, const float* l) {
  unsigned loff = (unsigned)(uintptr_t)l;
  unsigned long long ga = (unsigned long long)(uintptr_t)g;
  asm volatile("global_load_async_to_lds_b128 %0, %1, off th:TH_LOAD_NT"
               :: "v"(loff), "v"(ga) : "memory");
}

__device__ __forceinline__ void nt_store4(float* p, float a, float b, float c,
                                          float d) {
  v4f v = {a, b, c, d};
  __builtin_nontemporal_store(v, (v4f*)p);
}

// Issue one chunk (8 samples for 64 rays, all 6 arrays): 28 async b128
// instructions per wave. Global segments are 16B-aligned (s0 multiple of 8).
__device__ __forceinline__ void issue_chunk(
    float* buf, const float* gobj, const float* gmed, const float* gdir,
    const float* gbsc, const float* gden, const float* gdel, int s0, int tid) {
#pragma unroll
  for (int i = 0; i < 6; ++i) {            // 3-channel arrays: 96B per ray
    int idx = i * TPB + tid;               // 384 segments = 64 rays * 6
    int r   = idx / 6;
    int sg  = idx - r * 6;
    int go  = r * (S * 3) + s0 * 3 + sg * 4;   // floats
    int lo  = r * STRIDE3 + sg * 4;
    async_b128(gobj + go, buf + OFF_OBJ + lo);
    async_b128(gmed + go, buf + OFF_MED + lo);
    async_b128(gdir + go, buf + OFF_DIR + lo);
    async_b128(gbsc + go, buf + OFF_BSC + lo);
  }
#pragma unroll
  for (int i = 0; i < 2; ++i) {            // 1-channel arrays: 32B per ray
    int idx = i * TPB + tid;               // 128 segments = 64 rays * 2
    int r   = idx >> 1;
    int sg  = idx & 1;
    int go  = r * S + s0 + sg * 4;
    int lo  = r * STRIDE1 + sg * 4;
    async_b128(gden + go, buf + OFF_DEN + lo);
    async_b128(gdel + go, buf + OFF_DEL + lo);
  }
}

__global__ __launch_bounds__(TPB) void seathru_render_kernel(
    const float* __restrict__ obj, const float* __restrict__ med,
    const float* __restrict__ dirc, const float* __restrict__ bscc,
    const float* __restrict__ den, const float* __restrict__ del,
    float* __restrict__ out, int R) {
  extern __shared__ float smem[];
  float* bufA = smem;
  float* bufB = smem + BUF_FLOATS;

  const int tid  = threadIdx.x;
  const int ray0 = blockIdx.x * TPB;
  const int ray  = ray0 + tid;

  const float* gobj = obj  + (long)ray0 * S * 3;
  const float* gmed = med  + (long)ray0 * S * 3;
  const float* gdir = dirc + (long)ray0 * S * 3;
  const float* gbsc = bscc + (long)ray0 * S * 3;
  const float* gden = den  + (long)ray0 * S;
  const float* gdel = del  + (long)ray0 * S;

  // flat output-section offsets (floats), concat in reference return order
  const long oRGB = 0;
  const long oRST = 3L * R;
  const long oDIR = 6L * R;
  const long oBSK = 9L * R;
  const long oOT  = 12L * R;
  const long oOA  = oOT + (long)R * S;
  const long oOW  = oOA + (long)R * S;
  const long oDC0 = oOW + (long)R * S;
  const long oBC0 = oDC0 + 3L * R;
  const long oMSK = oBC0 + 3L * R;

  issue_chunk(bufA, gobj, gmed, gdir, gbsc, gden, gdel, 0, tid);

  float cum_obj = 0.f;                       // exclusive cumsum of delta*density
  float cumD[3] = {0.f, 0.f, 0.f};           // exclusive cumsum of dc*delta
  float cumB[3] = {0.f, 0.f, 0.f};           // exclusive cumsum of bc*delta
  float restored[3] = {0.f, 0.f, 0.f};
  float directA[3]  = {0.f, 0.f, 0.f};
  float backA[3]    = {0.f, 0.f, 0.f};
  float maskSum = 0.f;
  float dc0[3] = {0.f, 0.f, 0.f};
  float bc0[3] = {0.f, 0.f, 0.f};

  for (int c = 0; c < NCHUNK; ++c) {
    float* cur = (c & 1) ? bufB : bufA;
    float* nxt = (c & 1) ? bufA : bufB;
    if (c + 1 < NCHUNK) {
      issue_chunk(nxt, gobj, gmed, gdir, gbsc, gden, gdel, (c + 1) * CHUNK, tid);
      // <=56 outstanding; async loads retire in order, so waiting down to 28
      // guarantees chunk c fully landed while chunk c+1 stays in flight.
      asm volatile("s_wait_asynccnt 28" ::: "memory");
    } else {
      asm volatile("s_wait_asynccnt 0" ::: "memory");
    }
    __syncthreads();

    const float* po = cur + OFF_OBJ + tid * STRIDE3;
    const float* pm = cur + OFF_MED + tid * STRIDE3;
    const float* pd = cur + OFF_DIR + tid * STRIDE3;
    const float* pb = cur + OFF_BSC + tid * STRIDE3;
    const float* pn = cur + OFF_DEN + tid * STRIDE1;
    const float* pl = cur + OFF_DEL + tid * STRIDE1;

    if (c == 0) {
#pragma unroll
      for (int ch = 0; ch < 3; ++ch) { dc0[ch] = nn(pd[ch]); bc0[ch] = nn(pb[ch]); }
    }

    float Tb[CHUNK], Ab[CHUNK], Wb[CHUNK];
#pragma unroll
    for (int s = 0; s < CHUNK; ++s) {
      float delta = pl[s];
      float dens  = pn[s];
      float dd    = delta * dens;
      float T     = __expf(-cum_obj);        // exclusive transmittance
      float alpha = 1.f - __expf(-dd);
      float w     = T * alpha;
      Tb[s] = T; Ab[s] = alpha; Wb[s] = w;
      maskSum += w;
#pragma unroll
      for (int ch = 0; ch < 3; ++ch) {
        float orgb = nn(po[s * 3 + ch]);
        float mrgb = nn(pm[s * 3 + ch]);
        float dcv  = nn(pd[s * 3 + ch]);
        float bcv  = nn(pb[s * 3 + ch]);
        restored[ch] = fmaf(w, orgb, restored[ch]);
        float att = __expf(-cumD[ch]);                    // direct attenuation
        directA[ch] = fmaf(T * att * alpha, orgb, directA[ch]);
        float bsdd   = bcv * delta;
        float Bsc    = __expf(-cumB[ch]);                 // backscattering
        float malpha = 1.f - __expf(-bsdd);
        backA[ch] = fmaf(T * Bsc * malpha, mrgb, backA[ch]);
        cumD[ch] += dcv * delta;
        cumB[ch] += bsdd;
      }
      cum_obj += dd;
    }

    // per-sample outputs: write-once streams -> non-temporal b128 stores
    // (32B-aligned since the chunk base sample is a multiple of 8)
    long sBase = (long)ray * S + c * CHUNK;
    nt_store4(out + oOT + sBase,     Tb[0], Tb[1], Tb[2], Tb[3]);
    nt_store4(out + oOT + sBase + 4, Tb[4], Tb[5], Tb[6], Tb[7]);
    nt_store4(out + oOA + sBase,     Ab[0], Ab[1], Ab[2], Ab[3]);
    nt_store4(out + oOA + sBase + 4, Ab[4], Ab[5], Ab[6], Ab[7]);
    nt_store4(out + oOW + sBase,     Wb[0], Wb[1], Wb[2], Wb[3]);
    nt_store4(out + oOW + sBase + 4, Wb[4], Wb[5], Wb[6], Wb[7]);

    __syncthreads();  // cur may be overwritten by next iteration's async issue
  }

#pragma unroll
  for (int ch = 0; ch < 3; ++ch) {
    float dr = directA[ch], br = backA[ch];
    out[oRGB + (long)ray * 3 + ch] = clamp01(dr + br);
    out[oRST + (long)ray * 3 + ch] = clamp01(restored[ch]);
    out[oDIR + (long)ray * 3 + ch] = clamp01(dr);
    out[oBSK + (long)ray * 3 + ch] = clamp01(br);
    out[oDC0 + (long)ray * 3 + ch] = dc0[ch];
    out[oBC0 + (long)ray * 3 + ch] = bc0[ch];
  }
  out[oMSK + ray] = clamp01(maskSum);
}

extern "C" void kernel_launch(void* const* d_in, const int* in_sizes, int n_in,
                              void* d_out, int out_size, void* d_ws, size_t ws_size,
                              hipStream_t stream) {
  (void)n_in; (void)out_size; (void)d_ws; (void)ws_size;
  const float* obj  = (const float*)d_in[0];
  const float* med  = (const float*)d_in[1];
  const float* dirc = (const float*)d_in[2];
  const float* bscc = (const float*)d_in[3];
  const float* den  = (const float*)d_in[4];
  const float* del  = (const float*)d_in[5];
  float* out = (float*)d_out;

  const int R = in_sizes[4] / S;          // densities are [R,S,1]
  const int nblocks = R / TPB;            // 256 blocks of 64 threads (2 waves)
  const size_t shmem = (size_t)(2 * BUF_FLOATS) * sizeof(float);  // 65,536 B

  seathru_render_kernel<<<nblocks, TPB, shmem, stream>>>(
      obj, med, dirc, bscc, den, del, out, R);
}